// RoPE_19774029431219
// MI455X (gfx1250) — compile-verified
//
#include <hip/hip_runtime.h>
#include <math.h>

typedef float v4f __attribute__((ext_vector_type(4)));

#define S_LEN 8192
#define D_DIM 128
#define B_DIM 16
#define ROWS_PER_BLOCK 8
#define THREADS 256

// Issue a 16-byte async global->LDS copy for this lane.
// s_wait_dscnt 0 first: guarantees any in-flight ds_load from the buffer we
// are about to overwrite (previous iteration) has completed before the async
// engine can write LDS.
#define ASYNC_LD(ldsaddr, gaddr)                                              \
  asm volatile("s_wait_dscnt 0\n\t"                                           \
               "global_load_async_to_lds_b128 %0, %1, off"                    \
               :: "v"(ldsaddr), "v"(gaddr) : "memory")

__global__ __launch_bounds__(THREADS) void rope_async_kernel(
    const float* __restrict__ x, float* __restrict__ out) {
  // double buffer: 2 x 256 lanes x 16 B = 8 KB
  __shared__ __align__(16) float lbuf[2][THREADS * 4];

  const int tid = threadIdx.x;
  const int j   = tid & 31;   // float4 column within row: dims 4j..4j+3
  const int row = tid >> 5;   // row within block
  const int pos = blockIdx.x * ROWS_PER_BLOCK + row;

  // Rotation coefficients, computed once per thread, reused across 16 batches.
  // pairs p0 = 2j (dims 4j,4j+1), p1 = 2j+1 (dims 4j+2,4j+3)
  // inv_freq(p) = 10000^(-p/64) = exp2(-p * log2(10000)/64), f64 for accuracy.
  const double K = 13.287712379549449 / 64.0;  // log2(10000)/64
  const float inv0 = (float)exp2(-(double)(2 * j) * K);
  const float inv1 = (float)exp2(-(double)(2 * j + 1) * K);
  const float ang0 = (float)pos * inv0;
  const float ang1 = (float)pos * inv1;
  float s0, c0, s1, c1;
  sincosf(ang0, &s0, &c0);
  sincosf(ang1, &s1, &c1);

  // Per-lane addresses
  const size_t elem0 = (size_t)pos * D_DIM + (size_t)j * 4;  // batch-0 float idx
  const unsigned long long gstride =
      (unsigned long long)S_LEN * D_DIM * sizeof(float);     // 4 MiB per batch
  const unsigned long long ga = (unsigned long long)(uintptr_t)(x + elem0);

  const unsigned lds0 = (unsigned)(uintptr_t)(&lbuf[0][tid * 4]);
  const unsigned lds1 = (unsigned)(uintptr_t)(&lbuf[1][tid * 4]);

  // Prologue: batch 0 -> buffer 0
  ASYNC_LD(lds0, ga);

#pragma unroll
  for (int b = 0; b < B_DIM; ++b) {
    if (b + 1 < B_DIM) {
      // prefetch next batch into the other buffer, then wait for current
      ASYNC_LD(((b & 1) ? lds0 : lds1),
               ga + (unsigned long long)(b + 1) * gstride);
      asm volatile("s_wait_asynccnt 1" ::: "memory");
    } else {
      asm volatile("s_wait_asynccnt 0" ::: "memory");
    }

    // one ds_load_b128 from our own slot
    v4f in = *(const v4f*)(&lbuf[b & 1][tid * 4]);

    v4f o;
    o.x = in.x * c0 - in.y * s0;
    o.y = in.x * s0 + in.y * c0;
    o.z = in.z * c1 - in.w * s1;
    o.w = in.z * s1 + in.w * c1;

    // streaming 128-bit non-temporal store
    __builtin_nontemporal_store(
        o, (v4f*)(out + (size_t)b * (size_t)(S_LEN * D_DIM) + elem0));
  }
}

extern "C" void kernel_launch(void* const* d_in, const int* in_sizes, int n_in,
                              void* d_out, int out_size, void* d_ws,
                              size_t ws_size, hipStream_t stream) {
  (void)in_sizes; (void)n_in; (void)d_ws; (void)ws_size; (void)out_size;
  const float* x = (const float*)d_in[0];
  // d_in[1] (token_positions) is unused by the reference: it rotates by
  // arange(S), not by the provided positions.
  float* out = (float*)d_out;

  dim3 block(THREADS);
  dim3 grid(S_LEN / ROWS_PER_BLOCK);  // 1024 blocks x 256 threads
  rope_async_kernel<<<grid, block, 0, stream>>>(x, out);
}